// SIGN_5385888989320
// MI455X (gfx1250) — compile-verified
//
#include <hip/hip_runtime.h>
#include <hip/hip_bf16.h>

typedef __attribute__((ext_vector_type(2))) float v2f;
typedef __attribute__((ext_vector_type(8))) float v8f;

#define LN_EPS 1e-5f

// ---------------------------------------------------------------- utilities
__global__ void zero_f32(float* __restrict__ p, long long n) {
  long long i = (long long)blockIdx.x * blockDim.x + threadIdx.x;
  if (i < n) p[i] = 0.0f;
}

__global__ void degree_count(const int* __restrict__ dst, float* __restrict__ deg, int E) {
  int i = blockIdx.x * blockDim.x + threadIdx.x;
  if (i < E) atomicAdd(&deg[dst[i]], 1.0f);
}

__global__ void finalize_invdeg(float* __restrict__ deg, int n) {
  int i = blockIdx.x * blockDim.x + threadIdx.x;
  if (i < n) deg[i] = 1.0f / fmaxf(deg[i], 1.0f);
}

// scatter-add of 64-feature rows: 16 threads per edge, float4 each
__global__ void scatter_add64(const float* __restrict__ h, const int* __restrict__ src,
                              const int* __restrict__ dst, float* __restrict__ msg,
                              long long work) {
  long long i = (long long)blockIdx.x * blockDim.x + threadIdx.x;
  if (i >= work) return;
  int e = (int)(i >> 4);
  int f = (int)(i & 15) * 4;
  int s = src[e], d = dst[e];
  float4 v = *(const float4*)(h + (long long)s * 64 + f);
  float* mp = msg + (long long)d * 64 + f;
  atomicAdd(mp + 0, v.x);
  atomicAdd(mp + 1, v.y);
  atomicAdd(mp + 2, v.z);
  atomicAdd(mp + 3, v.w);
}

// -------------------------------------------- hop GEMM: (msg*invdeg) @ w[64,64]
// block: 128 threads = 4 waves, 16 rows; wave w owns 16 output columns.
__global__ __launch_bounds__(128) void hop_gemm64(
    const float* __restrict__ msg, const float* __restrict__ invdeg,
    const float* __restrict__ w,          // [64,64] row-major
    float* __restrict__ zout, int n) {
  __shared__ float As[16][68];            // 64 + 4 pad (stride%64==4 -> conflict-free)
  int tid = threadIdx.x;
  int row0 = blockIdx.x * 16;
  for (int i = tid; i < 16 * 64; i += 128) {
    int r = i >> 6, c = i & 63;
    int gr = min(row0 + r, n - 1);
    As[r][c] = msg[(long long)gr * 64 + c] * invdeg[gr];
  }
  __syncthreads();

  int wave = tid >> 5, lane = tid & 31;
  int half = lane >> 4, l16 = lane & 15;
  int cb = wave * 16;
  v8f acc = {};
  for (int kb = 0; kb < 64; kb += 4) {
    int k0 = kb + half * 2;
    v2f a, b;
    a.x = As[l16][k0];
    a.y = As[l16][k0 + 1];
    b.x = w[k0 * 64 + cb + l16];
    b.y = w[(k0 + 1) * 64 + cb + l16];
    acc = __builtin_amdgcn_wmma_f32_16x16x4_f32(false, a, false, b, (short)0, acc, false, false);
  }
#pragma unroll
  for (int v = 0; v < 8; ++v) {
    int r = row0 + v + half * 8;
    if (r < n) zout[(long long)r * 64 + cb + l16] = acc[v];
  }
}

// ---------------- fused MLP: h = relu(LN([z0|z1|z2|z3] @ w1 + b1)) @ w2 + b2
// block: 256 threads = 8 waves, one 16-row tile.
__global__ __launch_bounds__(256) void fused_mlp(
    const float* __restrict__ z0, const float* __restrict__ z1,
    const float* __restrict__ z2, const float* __restrict__ z3,
    const float* __restrict__ w1, const float* __restrict__ b1,   // [256,512], [512]
    const float* __restrict__ gamma, const float* __restrict__ beta,
    const float* __restrict__ w2, const float* __restrict__ b2,   // [512,64], [64]
    float* __restrict__ out, int n) {
  __shared__ float As[16][260];      // 256 + 4 pad
  __shared__ float Hs[16][516];      // 512 + 4 pad
  __shared__ float Ps[4][16][16];    // lin2 partial (k-half 1)

  int tid = threadIdx.x;
  int row0 = blockIdx.x * 16;

  // stage concatenated A tile [16,256]
  for (int i = tid; i < 16 * 256; i += 256) {
    int r = i >> 8, c = i & 255;
    int gr = min(row0 + r, n - 1);
    const float* zp = (c < 64) ? z0 : (c < 128) ? z1 : (c < 192) ? z2 : z3;
    As[r][c] = zp[(long long)gr * 64 + (c & 63)];
  }
  __syncthreads();

  int wave = tid >> 5, lane = tid & 31;
  int half = lane >> 4, l16 = lane & 15;

  // ---- GEMM1: wave owns 64 columns (4 n-tiles), K = 256
  int cb0 = wave * 64;
  v8f acc[4] = {};
  for (int kb = 0; kb < 256; kb += 4) {
    int k0 = kb + half * 2;
    v2f a;
    a.x = As[l16][k0];
    a.y = As[l16][k0 + 1];
    const float* w1p0 = w1 + (long long)k0 * 512;
    const float* w1p1 = w1 + (long long)(k0 + 1) * 512;
#pragma unroll
    for (int t = 0; t < 4; ++t) {
      int cb = cb0 + t * 16;
      v2f b;
      b.x = w1p0[cb + l16];
      b.y = w1p1[cb + l16];
      acc[t] = __builtin_amdgcn_wmma_f32_16x16x4_f32(false, a, false, b, (short)0, acc[t],
                                                     false, false);
    }
  }
#pragma unroll
  for (int t = 0; t < 4; ++t) {
    int cb = cb0 + t * 16;
#pragma unroll
    for (int v = 0; v < 8; ++v)
      Hs[v + half * 8][cb + l16] = acc[t][v] + b1[cb + l16];
  }
  __syncthreads();

  // ---- LayerNorm + ReLU (16 threads per row, butterfly reduce in half-wave)
  {
    int r = tid >> 4;          // row 0..15
    int sub = tid & 15;
    int c0 = sub * 32;
    float s = 0.f, ss = 0.f;
    for (int j = 0; j < 32; ++j) {
      float v = Hs[r][c0 + j];
      s += v;
      ss += v * v;
    }
    for (int m = 8; m >= 1; m >>= 1) {   // stays within 16-lane group
      s  += __shfl_xor(s, m, 32);
      ss += __shfl_xor(ss, m, 32);
    }
    float mu = s * (1.0f / 512.0f);
    float var = ss * (1.0f / 512.0f) - mu * mu;
    float rstd = rsqrtf(var + LN_EPS);
    for (int j = 0; j < 32; ++j) {
      int c = c0 + j;
      float v = (Hs[r][c] - mu) * rstd * gamma[c] + beta[c];
      Hs[r][c] = v > 0.0f ? v : 0.0f;
    }
  }
  __syncthreads();

  // ---- GEMM2: [16,512] @ [512,64]; wave -> (ntile = w&3, k-half = w>>2)
  int nt = wave & 3, kh = wave >> 2;
  int cb = nt * 16;
  v8f acc2 = {};
  int kend = kh * 256 + 256;
  for (int kb = kh * 256; kb < kend; kb += 4) {
    int k0 = kb + half * 2;
    v2f a, b;
    a.x = Hs[l16][k0];
    a.y = Hs[l16][k0 + 1];
    b.x = w2[(long long)k0 * 64 + cb + l16];
    b.y = w2[(long long)(k0 + 1) * 64 + cb + l16];
    acc2 = __builtin_amdgcn_wmma_f32_16x16x4_f32(false, a, false, b, (short)0, acc2,
                                                 false, false);
  }
  if (kh == 1) {
#pragma unroll
    for (int v = 0; v < 8; ++v) Ps[nt][v + half * 8][l16] = acc2[v];
  }
  __syncthreads();
  if (kh == 0) {
#pragma unroll
    for (int v = 0; v < 8; ++v) {
      int r = row0 + v + half * 8;
      float val = acc2[v] + Ps[nt][v + half * 8][l16] + b2[cb + l16];
      if (r < n) out[(long long)r * 64 + cb + l16] = val;
    }
  }
}

// ---------------------------------------------------------------- launcher
extern "C" void kernel_launch(void* const* d_in, const int* in_sizes, int n_in,
                              void* d_out, int out_size, void* d_ws, size_t ws_size,
                              hipStream_t stream) {
  const float* x      = (const float*)d_in[0];   // [N,64]
  const int*   edge   = (const int*)d_in[1];     // [2,E] (harness: integer -> int32)
  const float* w_prop = (const float*)d_in[2];   // [3,64,64]
  const float* lin1_w = (const float*)d_in[3];   // [256,512]
  const float* lin1_b = (const float*)d_in[4];   // [512]
  const float* gamma  = (const float*)d_in[5];   // [512]
  const float* beta   = (const float*)d_in[6];   // [512]
  const float* lin2_w = (const float*)d_in[7];   // [512,64]
  const float* lin2_b = (const float*)d_in[8];   // [64]
  float* out = (float*)d_out;

  const int N = in_sizes[0] / 64;
  const int E = in_sizes[1] / 2;
  const int* src = edge;
  const int* dst = edge + E;

  // workspace layout (floats): invdeg[N] | z1[N*64] | z2[N*64] | z3[N*64] | msg[N*64]
  float* ws = (float*)d_ws;
  float* invdeg = ws;
  float* z1 = ws + N;
  float* z2 = z1 + (long long)N * 64;
  float* z3 = z2 + (long long)N * 64;
  float* msg = z3 + (long long)N * 64;

  // degree -> inverse degree (graph is static; once per launch)
  zero_f32<<<(N + 255) / 256, 256, 0, stream>>>(invdeg, (long long)N);
  degree_count<<<(E + 255) / 256, 256, 0, stream>>>(dst, invdeg, E);
  finalize_invdeg<<<(N + 255) / 256, 256, 0, stream>>>(invdeg, N);

  long long sw = (long long)E * 16;             // 16 threads per edge
  int sgrid = (int)((sw + 255) / 256);
  long long mcnt = (long long)N * 64;
  int zgrid = (int)((mcnt + 255) / 256);
  int tgrid = (N + 15) / 16;                    // one 16-row tile per block

  const float* cur = x;
  float* zbuf[3] = {z1, z2, z3};
  for (int hop = 0; hop < 3; ++hop) {
    zero_f32<<<zgrid, 256, 0, stream>>>(msg, mcnt);
    scatter_add64<<<sgrid, 256, 0, stream>>>(cur, src, dst, msg, sw);
    hop_gemm64<<<tgrid, 128, 0, stream>>>(msg, invdeg, w_prop + hop * 64 * 64, zbuf[hop], N);
    cur = zbuf[hop];
  }

  fused_mlp<<<tgrid, 256, 0, stream>>>(x, z1, z2, z3, lin1_w, lin1_b, gamma, beta,
                                       lin2_w, lin2_b, out, N);
}